// localPatchClassifier_53893249630332
// MI455X (gfx1250) — compile-verified
//
#include <hip/hip_runtime.h>
#include <stdint.h>

typedef __attribute__((ext_vector_type(2))) float v2f;
typedef __attribute__((ext_vector_type(8))) float v8f;

#define WAY    5
#define NWQ    75
#define PQ     196          // query patches
#define SUP    980          // shot * patch_num
#define DIM    64
#define KTOP   3
#define NT     13           // ceil(PQ/16) query tiles
#define MT     62           // ceil(SUP/16) support tiles
#define NWAVES 8
#define LDS_ROW_DW 68       // 64 data dwords + 4 pad (bank-conflict-free ds_load_b64)
#define TILE_DW   (16 * LDS_ROW_DW)

// branchless sorted-descending top-3 insertion (5 VALU ops, no EXEC churn)
__device__ __forceinline__ void top3_insert(float v, float& t0, float& t1, float& t2) {
    float a0 = fmaxf(t0, v);
    float b0 = fminf(t0, v);
    float a1 = fmaxf(t1, b0);
    float b1 = fminf(t1, b0);
    float a2 = fmaxf(t2, b1);
    t0 = a0; t1 = a1; t2 = a2;
}

__global__ __launch_bounds__(256)
void lpc_wmma_kernel(const float* __restrict__ qf,
                     const float* __restrict__ sf,
                     float* __restrict__ out) {
    __shared__ __attribute__((aligned(16))) float smem[2 * TILE_DW];
    __shared__ float red[NWAVES];

    const int tid  = threadIdx.x;
    const int lane = tid & 31;
    const int wid  = tid >> 5;          // wave 0..7
    const int bqw  = blockIdx.x;        // (b*NWQ + q)*WAY + w
    const int w    = bqw % WAY;
    const int bq   = bqw / WAY;
    const int b    = bq / NWQ;

    const float* qbase = qf + (size_t)bq * PQ * DIM;
    const float* sbase = sf + ((size_t)(b * WAY + w)) * SUP * DIM;

    const int half = lane >> 4;         // fp32-WMMA lane-half
    const int r16  = lane & 15;
    const int koff = half * 2;          // per-lane k offset inside K=4 step

    // ---- per-wave query tiles (WMMA B operand, kept in registers) ----
    const int tA   = wid;               // n-tiles 0..7
    const int tB   = wid + NWAVES;      // n-tiles 8..12 (waves 0..4 only)
    const bool hasB = (tB < NT);

    const int qpA = tA * 16 + r16;
    const int qpB = tB * 16 + r16;
    const int qcA = qpA < PQ ? qpA : PQ - 1;
    const int qcB = qpB < PQ ? qpB : PQ - 1;

    v2f qbA[16], qbB[16];
    {
        const float* qa = qbase + (size_t)qcA * DIM;
#pragma unroll
        for (int kk = 0; kk < 16; ++kk)
            qbA[kk] = *(const v2f*)(qa + 4 * kk + koff);
    }
    if (hasB) {
        const float* qb_ = qbase + (size_t)qcB * DIM;
#pragma unroll
        for (int kk = 0; kk < 16; ++kk)
            qbB[kk] = *(const v2f*)(qb_ + 4 * kk + koff);
    }

    // ---- async staging: wave `wid` copies rows 2*wid, 2*wid+1 of each tile ----
    const int srow   = 2 * wid + half;  // row 0..15 of the 16-row tile
    const int schunk = r16;             // 16-byte chunk within the 256B row

    auto issue_tile = [&](int mt, int buf) {
        int mrow = mt * 16 + srow;
        mrow = mrow < SUP ? mrow : SUP - 1;                    // clamp pad rows
        const float* g = sbase + (size_t)mrow * DIM + schunk * 4;
        uint32_t laddr = (uint32_t)(uintptr_t)(&smem[buf * TILE_DW + srow * LDS_ROW_DW])
                       + (uint32_t)(schunk * 16);
        asm volatile("global_load_async_to_lds_b128 %0, %1, off"
                     :: "v"(laddr), "v"((unsigned long long)(uintptr_t)g)
                     : "memory");
    };

    auto wait_async_le1 = []() {
#if __has_builtin(__builtin_amdgcn_s_wait_asynccnt)
        __builtin_amdgcn_s_wait_asynccnt(1);
#else
        asm volatile("s_wait_asynccnt 0x1" ::: "memory");
#endif
    };

    float a0 = -__builtin_inff(), a1 = -__builtin_inff(), a2 = -__builtin_inff();
    float b0 = -__builtin_inff(), b1 = -__builtin_inff(), b2 = -__builtin_inff();

    issue_tile(0, 0);                                          // prologue prefetch

    for (int mt = 0; mt < MT; ++mt) {
        const int cur = mt & 1;
        const int nxt = (mt + 1 < MT) ? (mt + 1) : (MT - 1);   // dummy re-issue on last
        issue_tile(nxt, (mt + 1) & 1);                         // prefetch next tile
        wait_async_le1();                                      // tile `mt` resident
        __syncthreads();                                       // all waves staged

        // gather this lane's A-operand slices from LDS (conflict-free ds_load_b64)
        const float* tile = &smem[cur * TILE_DW] + r16 * LDS_ROW_DW + koff;
        v2f sa[16];
#pragma unroll
        for (int kk = 0; kk < 16; ++kk)
            sa[kk] = *(const v2f*)(tile + 4 * kk);

        // Sᵀ tile = sup(16x64) x queryᵀ(64x16): 16 chained K=4 fp32 WMMAs per n-tile
        v8f cA = {};
#pragma unroll
        for (int kk = 0; kk < 16; ++kk)
            cA = __builtin_amdgcn_wmma_f32_16x16x4_f32(
                    false, sa[kk], false, qbA[kk], (short)0, cA, false, false);
        v8f cB = {};
        if (hasB) {
#pragma unroll
            for (int kk = 0; kk < 16; ++kk)
                cB = __builtin_amdgcn_wmma_f32_16x16x4_f32(
                        false, sa[kk], false, qbB[kk], (short)0, cB, false, false);
        }

        // lane owns rows m = mt*16 + half*8 + j for its query column; mask pad rows
        const int mlimit = SUP - mt * 16;                      // rows valid this tile
#pragma unroll
        for (int j = 0; j < 8; ++j) {
            const bool ok = (half * 8 + j) < mlimit;
            top3_insert(ok ? cA[j] : -__builtin_inff(), a0, a1, a2);
        }
        if (hasB) {
#pragma unroll
            for (int j = 0; j < 8; ++j) {
                const bool ok = (half * 8 + j) < mlimit;
                top3_insert(ok ? cB[j] : -__builtin_inff(), b0, b1, b2);
            }
        }
        __syncthreads();                                       // tile consumed
    }

    // ---- merge the two half-lane top-3 lists, sum, reduce, write ----
    float s;
    {
        float p0 = __shfl_xor(a0, 16, 32);
        float p1 = __shfl_xor(a1, 16, 32);
        float p2 = __shfl_xor(a2, 16, 32);
        top3_insert(p0, a0, a1, a2);
        top3_insert(p1, a0, a1, a2);
        top3_insert(p2, a0, a1, a2);
        float sa_ = a0 + a1 + a2;
        if (qpA >= PQ || half) sa_ = 0.0f;                     // pad patch / dup half
        s = sa_;
    }
    if (hasB) {
        float p0 = __shfl_xor(b0, 16, 32);
        float p1 = __shfl_xor(b1, 16, 32);
        float p2 = __shfl_xor(b2, 16, 32);
        top3_insert(p0, b0, b1, b2);
        top3_insert(p1, b0, b1, b2);
        top3_insert(p2, b0, b1, b2);
        float sb_ = b0 + b1 + b2;
        if (qpB >= PQ || half) sb_ = 0.0f;
        s += sb_;
    }
#pragma unroll
    for (int m = 1; m < 16; m <<= 1)
        s += __shfl_xor(s, m, 32);
    if (lane == 0) red[wid] = s;
    __syncthreads();
    if (tid == 0) {
        float t = 0.0f;
#pragma unroll
        for (int i = 0; i < NWAVES; ++i) t += red[i];
        out[bqw] = t * (1.0f / (float)(PQ * KTOP));
    }
}

extern "C" void kernel_launch(void* const* d_in, const int* in_sizes, int n_in,
                              void* d_out, int out_size, void* d_ws, size_t ws_size,
                              hipStream_t stream) {
    const float* qf = (const float*)d_in[0];   // [b, 75, 196, 64] fp32
    const float* sf = (const float*)d_in[1];   // [b, 5, 5, 196, 64] fp32
    float* out = (float*)d_out;                // [b*75, 5] fp32

    const int b    = in_sizes[0] / (NWQ * PQ * DIM);
    const int nout = b * NWQ * WAY;            // one workgroup per (b,q,w)

    lpc_wmma_kernel<<<nout, 256, 0, stream>>>(qf, sf, out);
}